// GATEncoder_18519898980763
// MI455X (gfx1250) — compile-verified
//
#include <hip/hip_runtime.h>
#include <math.h>
#include <stdint.h>

// ---------------------------------------------------------------------------
// GAT encoder for MI455X (gfx1250, wave32).
// GEMMs use V_WMMA_F32_16X16X4_F32 (fp32 tensor path); W is staged into LDS
// with GLOBAL_LOAD_ASYNC_TO_LDS_B128 (ASYNCcnt) once per block. Edge
// softmax/aggregation and pooling use L2-resident float atomics (23.3 TB/s
// HBM, 192MB L2 -> memory/atomic bound; dense FLOPs are negligible).
// ---------------------------------------------------------------------------

typedef __attribute__((ext_vector_type(2))) float v2f;
typedef __attribute__((ext_vector_type(8))) float v8f;

// ---------------- WMMA GEMM: Y[N,128] = X[N,128] @ W[128,128] ---------------
// 256 threads = 8 waves; each wave computes a 16x128 tile (8 col-tiles of 16).
// W staged in LDS (64KB) via async-to-LDS. K-loop: 32 steps of K=4.
__global__ __launch_bounds__(256) void gemm_xw_wmma(
    const float* __restrict__ X, const float* __restrict__ W,
    float* __restrict__ Y, int n)
{
  __shared__ float sW[128 * 128];

  // Async copy W -> LDS: 256 threads x 16B x 16 steps = 64KB (ASYNCcnt path).
  // The LDS address operand is derived from the sW pointer itself: the low 32
  // bits of a generic LDS address are the LDS byte offset (ISA aperture rule),
  // and the ptrtoint capture makes the "memory"-clobbering asm a visible
  // writer of sW so the ds_load reads below are not folded to undef.
  {
    unsigned ldsbase  = (unsigned)(uintptr_t)(void*)sW;
    unsigned lds_off  = ldsbase + threadIdx.x * 16;   // 16B per thread
    const float* wsrc = W + threadIdx.x * 4;
#pragma unroll
    for (int i = 0; i < 16; ++i) {
      asm volatile("global_load_async_to_lds_b128 %0, %1, off"
                   :
                   : "v"(lds_off + i * 4096u), "v"(wsrc + i * 1024)
                   : "memory");
    }
    asm volatile("s_wait_asynccnt 0" ::: "memory");
  }
  __syncthreads();

  const int wave = threadIdx.x >> 5;
  const int lane = threadIdx.x & 31;
  const int half = lane >> 4;     // 0: K=0,1  1: K=2,3 (A layout)
  const int m    = lane & 15;     // row within tile (A) / col (B,D)
  const int rowTile = (blockIdx.x * 8 + wave) * 16;

  // Clamp out-of-range rows to a valid one: in 16x16x4 WMMA an A row feeds
  // only its own D row, and out-of-range D rows are never stored, so no
  // zero-fill (and no per-iteration EXEC branch) is needed.
  int row = rowTile + m;
  if (row >= n) row = n - 1;
  const float* xrow = X + (size_t)row * 128 + 2 * half;

  v8f acc[8] = {};

  for (int k0 = 0; k0 < 128; k0 += 4) {
    v2f a;
    a.x = xrow[k0 + 0];
    a.y = xrow[k0 + 1];
#pragma unroll
    for (int t = 0; t < 8; ++t) {
      v2f b;
      b.x = sW[(k0 + 2 * half + 0) * 128 + t * 16 + m];
      b.y = sW[(k0 + 2 * half + 1) * 128 + t * 16 + m];
      acc[t] = __builtin_amdgcn_wmma_f32_16x16x4_f32(
          false, a, false, b, (short)0, acc[t], false, false);
    }
  }

  // D layout: lanes 0-15 -> M = r, lanes 16-31 -> M = r + 8, col = lane&15
#pragma unroll
  for (int t = 0; t < 8; ++t) {
    const int col = t * 16 + m;
#pragma unroll
    for (int r = 0; r < 8; ++r) {
      const int orow = rowTile + half * 8 + r;
      if (orow < n) Y[(size_t)orow * 128 + col] = acc[t][r];
    }
  }
}

// ---------------- helpers ---------------------------------------------------
__device__ __forceinline__ void atomic_max_float(float* addr, float val) {
  if (val >= 0.0f) atomicMax((int*)addr, __float_as_int(val));
  else             atomicMin((unsigned int*)addr, __float_as_uint(val));
}

__global__ void fill_kernel(float* __restrict__ p, float v, int n) {
  int i = blockIdx.x * blockDim.x + threadIdx.x;
  if (i < n) p[i] = v;
}

// per-node attention coefficients: asrc[n,h] = <xl[n,h,:], a_src[h,:]>
__global__ void attn_kernel(const float* __restrict__ XL,
                            const float* __restrict__ avs,
                            const float* __restrict__ avd,
                            float* __restrict__ asrc, float* __restrict__ adst,
                            int n, int H, int hs, int C) {
  int id = blockIdx.x * blockDim.x + threadIdx.x;
  if (id >= (n << hs)) return;
  int node = id >> hs, h = id & (H - 1);
  const float* xp = XL + (size_t)node * 128 + h * C;
  float ss = 0.f, sd = 0.f;
  for (int c = 0; c < C; ++c) {
    float v = xp[c];
    ss += v * avs[h * C + c];
    sd += v * avd[h * C + c];
  }
  asrc[id] = ss;
  adst[id] = sd;
}

__device__ __forceinline__ void edge_ends(const long long* __restrict__ es,
                                          const long long* __restrict__ ed,
                                          int e, int E, int& src, int& dst) {
  if (e < E) { src = (int)es[e]; dst = (int)ed[e]; }
  else       { src = dst = e - E; }   // self loop
}

// pass 1: segment max of leaky_relu(asrc[src]+adst[dst]) over dst
__global__ void edge_max_kernel(const long long* __restrict__ es,
                                const long long* __restrict__ ed, int E, int n,
                                const float* __restrict__ asrc,
                                const float* __restrict__ adst,
                                float* __restrict__ M, int H, int hs) {
  int id = blockIdx.x * blockDim.x + threadIdx.x;
  int total = (E + n) << hs;
  if (id >= total) return;
  int e = id >> hs, h = id & (H - 1);
  int src, dst; edge_ends(es, ed, e, E, src, dst);
  float s = asrc[((size_t)src << hs) + h] + adst[((size_t)dst << hs) + h];
  s = s > 0.f ? s : 0.2f * s;
  atomic_max_float(&M[((size_t)dst << hs) + h], s);
}

// pass 2: ex = exp(s - m[dst]); denom[dst] += ex; store ex per edge
__global__ void edge_exp_kernel(const long long* __restrict__ es,
                                const long long* __restrict__ ed, int E, int n,
                                const float* __restrict__ asrc,
                                const float* __restrict__ adst,
                                const float* __restrict__ M,
                                float* __restrict__ EXB,
                                float* __restrict__ DEN, int H, int hs) {
  int id = blockIdx.x * blockDim.x + threadIdx.x;
  int total = (E + n) << hs;
  if (id >= total) return;
  int e = id >> hs, h = id & (H - 1);
  int src, dst; edge_ends(es, ed, e, E, src, dst);
  float s = asrc[((size_t)src << hs) + h] + adst[((size_t)dst << hs) + h];
  s = s > 0.f ? s : 0.2f * s;
  float ex = __expf(s - M[((size_t)dst << hs) + h]);
  EXB[id] = ex;
  atomicAdd(&DEN[((size_t)dst << hs) + h], ex);
}

// pass 3: AGG[dst,:] += alpha * xl[src,:]   (4 channels / thread, float4 gather)
__global__ void edge_agg_kernel(const long long* __restrict__ es,
                                const long long* __restrict__ ed, int E, int n,
                                const float* __restrict__ XL,
                                const float* __restrict__ EXB,
                                const float* __restrict__ DEN,
                                float* __restrict__ AGG, int H, int hs, int C) {
  int id = blockIdx.x * blockDim.x + threadIdx.x;
  int total = (E + n) * 32;
  if (id >= total) return;
  int e = id >> 5, q = id & 31;
  int src, dst; edge_ends(es, ed, e, E, src, dst);
  int h = (q * 4) / C;            // C is 32 or 128 -> strength-reduced
  float alpha = EXB[((size_t)e << hs) + h] /
                (DEN[((size_t)dst << hs) + h] + 1e-16f);
  float4 v = *(const float4*)(XL + (size_t)src * 128 + q * 4);
  float* o = AGG + (size_t)dst * 128 + q * 4;
  atomicAdd(o + 0, v.x * alpha);
  atomicAdd(o + 1, v.y * alpha);
  atomicAdd(o + 2, v.z * alpha);
  atomicAdd(o + 3, v.w * alpha);
}

// BN statistics: per-channel sum / sumsq (128 threads = 1 channel each,
// 256 rows per block, coalesced across channels)
__global__ void bn_stat_kernel(const float* __restrict__ A,
                               float* __restrict__ sum, float* __restrict__ sq,
                               int n) {
  int c = threadIdx.x;                 // 0..127
  int r0 = blockIdx.x * 256;
  int r1 = r0 + 256; if (r1 > n) r1 = n;
  float s = 0.f, q = 0.f;
  for (int r = r0; r < r1; ++r) {
    float v = A[(size_t)r * 128 + c];
    s += v; q += v * v;
  }
  atomicAdd(&sum[c], s);
  atomicAdd(&sq[c], q);
}

// BN apply (+ optional ELU). (Bias b cancels inside BN; skipped.)
__global__ void bn_apply_kernel(const float* __restrict__ A,
                                const float* __restrict__ sum,
                                const float* __restrict__ sq,
                                const float* __restrict__ g,
                                const float* __restrict__ be,
                                float* __restrict__ Y, int n, int do_elu) {
  int id = blockIdx.x * blockDim.x + threadIdx.x;
  if (id >= n * 128) return;
  int c = id & 127;
  float invn = 1.0f / (float)n;
  float mu  = sum[c] * invn;
  float var = sq[c] * invn - mu * mu;
  float v = (A[id] - mu) * rsqrtf(var + 1e-5f) * g[c] + be[c];
  if (do_elu) v = v > 0.f ? v : (__expf(v) - 1.f);
  Y[id] = v;
}

// ---------------- pooling ---------------------------------------------------
__global__ void pool_count_kernel(const long long* __restrict__ batch,
                                  float* __restrict__ cnt, int n) {
  int i = blockIdx.x * blockDim.x + threadIdx.x;
  if (i < n) atomicAdd(&cnt[(int)batch[i]], 1.0f);
}

__global__ void pool_acc_kernel(const float* __restrict__ Hf,
                                const long long* __restrict__ batch,
                                float* __restrict__ psum,
                                float* __restrict__ pmax, int n) {
  int id = blockIdx.x * blockDim.x + threadIdx.x;
  if (id >= n * 32) return;
  int node = id >> 5, q = id & 31;
  int g = (int)batch[node];
  float4 v = *(const float4*)(Hf + (size_t)node * 128 + q * 4);
  float* ps = psum + (size_t)g * 128 + q * 4;
  float* pm = pmax + (size_t)g * 128 + q * 4;
  atomicAdd(ps + 0, v.x); atomicAdd(ps + 1, v.y);
  atomicAdd(ps + 2, v.z); atomicAdd(ps + 3, v.w);
  atomic_max_float(pm + 0, v.x); atomic_max_float(pm + 1, v.y);
  atomic_max_float(pm + 2, v.z); atomic_max_float(pm + 3, v.w);
}

__global__ void pool_fin_kernel(const float* __restrict__ psum,
                                const float* __restrict__ pmax,
                                const float* __restrict__ cnt,
                                float* __restrict__ out, int G) {
  int id = blockIdx.x * blockDim.x + threadIdx.x;
  if (id >= G * 128) return;
  int g = id >> 7, c = id & 127;
  out[(size_t)g * 256 + c]       = psum[id] / fmaxf(cnt[g], 1.0f);
  out[(size_t)g * 256 + 128 + c] = pmax[id];
}

// ---------------------------------------------------------------------------
extern "C" void kernel_launch(void* const* d_in, const int* in_sizes, int n_in,
                              void* d_out, int out_size, void* d_ws, size_t ws_size,
                              hipStream_t stream) {
  const float*     x     = (const float*)d_in[0];
  const long long* ei    = (const long long*)d_in[1];   // [2,E] int64
  const long long* batch = (const long long*)d_in[2];   // [N] int64
  const float* W1  = (const float*)d_in[3];
  const float* as1 = (const float*)d_in[4];
  const float* ad1 = (const float*)d_in[5];
  const float* W2  = (const float*)d_in[7];
  const float* as2 = (const float*)d_in[8];
  const float* ad2 = (const float*)d_in[9];
  const float* W3  = (const float*)d_in[11];
  const float* as3 = (const float*)d_in[12];
  const float* ad3 = (const float*)d_in[13];
  const float* g1  = (const float*)d_in[15];
  const float* be1 = (const float*)d_in[16];
  const float* g2  = (const float*)d_in[17];
  const float* be2 = (const float*)d_in[18];
  const float* g3  = (const float*)d_in[19];
  const float* be3 = (const float*)d_in[20];

  const int N  = in_sizes[0] / 128;
  const int E  = in_sizes[1] / 2;
  const int G  = out_size / 256;
  const int EP = E + N;               // edges incl. self-loops
  const long long* esrc = ei;
  const long long* edst = ei + E;

  // ---- carve workspace ----
  size_t off = 0;
  auto carve = [&](size_t nfloats) -> float* {
    float* p = (float*)((char*)d_ws + off);
    off += ((nfloats * sizeof(float) + 255) / 256) * 256;
    return p;
  };
  float* XL   = carve((size_t)N * 128);   // GEMM output (per layer)
  float* AGG  = carve((size_t)N * 128);   // aggregated messages (pre-BN)
  float* Hbuf = carve((size_t)N * 128);   // post BN/ELU features
  float* ASRC = carve((size_t)N * 4);
  float* ADST = carve((size_t)N * 4);
  float* M    = carve((size_t)N * 4);
  float* DEN  = carve((size_t)N * 4);
  float* EXB  = carve((size_t)EP * 4);
  float* BNS  = carve(128);
  float* BNQ  = carve(128);
  float* PSUM = carve((size_t)G * 128);
  float* PMAX = carve((size_t)G * 128);
  float* PCNT = carve((size_t)G);
  (void)ws_size; (void)n_in;

  auto blocks = [](long long work, int tpb) -> int {
    return (int)((work + tpb - 1) / tpb);
  };

  auto run_layer = [&](const float* hin, const float* W, const float* avs,
                       const float* avd, const float* g, const float* be,
                       int H, int hs, int C, int do_elu) {
    hipMemsetAsync(AGG, 0, (size_t)N * 128 * sizeof(float), stream);
    hipMemsetAsync(DEN, 0, (size_t)N * H * sizeof(float), stream);
    hipMemsetAsync(BNS, 0, 128 * sizeof(float), stream);
    hipMemsetAsync(BNQ, 0, 128 * sizeof(float), stream);
    fill_kernel<<<blocks((long long)N * H, 256), 256, 0, stream>>>(
        M, -INFINITY, N * H);

    gemm_xw_wmma<<<blocks(N, 128), 256, 0, stream>>>(hin, W, XL, N);
    attn_kernel<<<blocks((long long)N * H, 256), 256, 0, stream>>>(
        XL, avs, avd, ASRC, ADST, N, H, hs, C);
    edge_max_kernel<<<blocks((long long)EP * H, 256), 256, 0, stream>>>(
        esrc, edst, E, N, ASRC, ADST, M, H, hs);
    edge_exp_kernel<<<blocks((long long)EP * H, 256), 256, 0, stream>>>(
        esrc, edst, E, N, ASRC, ADST, M, EXB, DEN, H, hs);
    edge_agg_kernel<<<blocks((long long)EP * 32, 256), 256, 0, stream>>>(
        esrc, edst, E, N, XL, EXB, DEN, AGG, H, hs, C);
    bn_stat_kernel<<<blocks(N, 256), 128, 0, stream>>>(AGG, BNS, BNQ, N);
    bn_apply_kernel<<<blocks((long long)N * 128, 256), 256, 0, stream>>>(
        AGG, BNS, BNQ, g, be, Hbuf, N, do_elu);
  };

  // layer 1: x -> Hbuf (4 heads x 32)
  run_layer(x,    W1, as1, ad1, g1, be1, 4, 2, 32, 1);
  // layer 2: Hbuf -> Hbuf
  run_layer(Hbuf, W2, as2, ad2, g2, be2, 4, 2, 32, 1);
  // layer 3: Hbuf -> Hbuf (1 head x 128, no ELU)
  run_layer(Hbuf, W3, as3, ad3, g3, be3, 1, 0, 128, 0);

  // pooling
  hipMemsetAsync(PSUM, 0, (size_t)G * 128 * sizeof(float), stream);
  hipMemsetAsync(PCNT, 0, (size_t)G * sizeof(float), stream);
  fill_kernel<<<blocks((long long)G * 128, 256), 256, 0, stream>>>(
      PMAX, -INFINITY, G * 128);
  pool_count_kernel<<<blocks(N, 256), 256, 0, stream>>>(batch, PCNT, N);
  pool_acc_kernel<<<blocks((long long)N * 32, 256), 256, 0, stream>>>(
      Hbuf, batch, PSUM, PMAX, N);
  pool_fin_kernel<<<blocks((long long)G * 128, 256), 256, 0, stream>>>(
      PSUM, PMAX, PCNT, (float*)d_out, G);
}